// DraiModel_48584670052891
// MI455X (gfx1250) — compile-verified
//
#include <hip/hip_runtime.h>
#include <hip/hip_bf16.h>
#include <math.h>

#define H      128
#define VOCAB  16384
#define NCODES 160000
#define SEG    8192
#define L      8193          // SEG + 1
#define MPAD   8208          // 513 * 16
#define MTILES 513
#define NTILES 1024          // VOCAB / 16

typedef float v2f __attribute__((ext_vector_type(2)));
typedef float v8f __attribute__((ext_vector_type(8)));

// ---------------------------------------------------------------------------
// Kernel 1: sum_emb[v] = sum_h emb[v][h].  One wave (32 lanes) per vocab row.
// ---------------------------------------------------------------------------
__global__ __launch_bounds__(256) void k_embsum(const float* __restrict__ emb,
                                                float* __restrict__ sum_emb) {
  int gtid = blockIdx.x * blockDim.x + threadIdx.x;
  int row  = gtid >> 5;
  int lane = gtid & 31;
  if (row >= VOCAB) return;
  const float4 v = *(const float4*)(emb + (size_t)row * H + lane * 4);
  float p = (v.x + v.y) + (v.z + v.w);
  #pragma unroll
  for (int off = 16; off > 0; off >>= 1) p += __shfl_down(p, off, 32);
  if (lane == 0) sum_emb[row] = p;
}

// ---------------------------------------------------------------------------
// Kernel 2: scalars[s] = sum_{i in [visits[s], visits[s+1])} sum_emb[codes[i]]
// ---------------------------------------------------------------------------
__global__ __launch_bounds__(256) void k_segsum(const int* __restrict__ codes,
                                                const int* __restrict__ visits,
                                                const float* __restrict__ sum_emb,
                                                float* __restrict__ scalars) {
  int s = blockIdx.x * blockDim.x + threadIdx.x;
  if (s >= SEG) return;
  int b = (s == 0) ? 0 : visits[s];
  int e = visits[s + 1];
  float acc = 0.f;
  for (int i = b; i < e; ++i) acc += sum_emb[codes[i]];
  scalars[s] = acc;
}

// ---------------------------------------------------------------------------
// Kernel 3: serial GRU scan. One block, 128 threads; thread j owns hidden
// unit j and computes all three gate dots (rows j, j+128, j+256 of w_hh).
// h double-buffered in LDS -> one barrier per step. Also zero-pads rows
// L..MPAD-1 of the output buffer for the WMMA GEMM.
// ---------------------------------------------------------------------------
__global__ __launch_bounds__(128) void k_gru(const float* __restrict__ scalars,
                                             const float* __restrict__ w_ih,
                                             const float* __restrict__ w_hh,
                                             const float* __restrict__ b_ih,
                                             const float* __restrict__ b_hh,
                                             float* __restrict__ outbuf /* MPAD x H */) {
  __shared__ float hbuf[2][H];
  int j = threadIdx.x;  // 0..127

  // rowsum(w_ih) for this thread's three gate rows
  float wr_s = 0.f, wz_s = 0.f, wn_s = 0.f;
  const float* wi_r = w_ih + (size_t)j * H;
  const float* wi_z = w_ih + (size_t)(j + H) * H;
  const float* wi_n = w_ih + (size_t)(j + 2 * H) * H;
  for (int k = 0; k < H; k += 4) {
    wr_s += (wi_r[k] + wi_r[k+1]) + (wi_r[k+2] + wi_r[k+3]);
    wz_s += (wi_z[k] + wi_z[k+1]) + (wi_z[k+2] + wi_z[k+3]);
    wn_s += (wi_n[k] + wi_n[k+1]) + (wi_n[k+2] + wi_n[k+3]);
  }
  const float bir = b_ih[j], biz = b_ih[j + H], bin_ = b_ih[j + 2 * H];
  const float bhr = b_hh[j], bhz = b_hh[j + H], bhn = b_hh[j + 2 * H];
  const float* wh_r = w_hh + (size_t)j * H;
  const float* wh_z = w_hh + (size_t)(j + H) * H;
  const float* wh_n = w_hh + (size_t)(j + 2 * H) * H;

  // zero the GEMM padding rows [L, MPAD)
  for (int idx = j; idx < (MPAD - L) * H; idx += 128) outbuf[(size_t)L * H + idx] = 0.f;

  hbuf[0][j] = 0.f;
  hbuf[1][j] = 0.f;
  __syncthreads();

  int cur = 0;
  for (int t = 0; t < L; ++t) {
    const float scal = (t < SEG) ? scalars[t] : 0.f;
    const float* hp = hbuf[cur];

    float ar0 = 0.f, ar1 = 0.f, az0 = 0.f, az1 = 0.f, an0 = 0.f, an1 = 0.f;
    for (int k = 0; k < H; k += 4) {
      float4 hv = *(const float4*)(hp + k);
      float4 wr = *(const float4*)(wh_r + k);
      float4 wz = *(const float4*)(wh_z + k);
      float4 wn = *(const float4*)(wh_n + k);
      ar0 = fmaf(wr.x, hv.x, ar0); ar1 = fmaf(wr.y, hv.y, ar1);
      ar0 = fmaf(wr.z, hv.z, ar0); ar1 = fmaf(wr.w, hv.w, ar1);
      az0 = fmaf(wz.x, hv.x, az0); az1 = fmaf(wz.y, hv.y, az1);
      az0 = fmaf(wz.z, hv.z, az0); az1 = fmaf(wz.w, hv.w, az1);
      an0 = fmaf(wn.x, hv.x, an0); an1 = fmaf(wn.y, hv.y, an1);
      an0 = fmaf(wn.z, hv.z, an0); an1 = fmaf(wn.w, hv.w, an1);
    }
    const float gh_r = bhr + ar0 + ar1;
    const float gh_z = bhz + az0 + az1;
    const float gh_n = bhn + an0 + an1;
    const float gi_r = fmaf(scal, wr_s, bir);
    const float gi_z = fmaf(scal, wz_s, biz);
    const float gi_n = fmaf(scal, wn_s, bin_);

    const float r = 1.f / (1.f + __expf(-(gi_r + gh_r)));
    const float z = 1.f / (1.f + __expf(-(gi_z + gh_z)));
    const float n = tanhf(fmaf(r, gh_n, gi_n));
    const float hold = hp[j];
    const float hnew = fmaf(z, hold - n, n);  // (1-z)*n + z*h

    hbuf[cur ^ 1][j] = hnew;
    outbuf[(size_t)t * H + j] = hnew;
    cur ^= 1;
    __syncthreads();
  }
}

// ---------------------------------------------------------------------------
// Kernel 4: logits = out @ w_out^T + b_out via V_WMMA_F32_16X16X4_F32.
// 256 threads = 8 waves per block; block owns a 16-row tile, waves stripe
// over 1024 column tiles. A (16x128) preloaded into 64 VGPRs/lane.
// f32 operand layout (ISA 7.12.2): lane = M/N index (mod 16), lane-half
// selects K pair {0,1}/{2,3} -> float2 loads at offset 4*kk + 2*hi.
// ---------------------------------------------------------------------------
__global__ __launch_bounds__(256) void k_gemm(const float* __restrict__ A,     /* MPAD x H */
                                              const float* __restrict__ W,     /* VOCAB x H */
                                              const float* __restrict__ bias,  /* VOCAB */
                                              float* __restrict__ out /* L x VOCAB */) {
  const int wave = threadIdx.x >> 5;
  const int lane = threadIdx.x & 31;
  const int n    = lane & 15;
  const int hi   = lane >> 4;
  const int rb   = blockIdx.x;  // 0..MTILES-1

  const float* ap = A + (size_t)(rb * 16 + n) * H + 2 * hi;
  v2f a[32];
  #pragma unroll
  for (int kk = 0; kk < 32; ++kk) a[kk] = *(const v2f*)(ap + kk * 4);

  for (int ct = wave; ct < NTILES; ct += 8) {
    const int col_base = ct * 16;
    const float* bp = W + (size_t)(col_base + n) * H + 2 * hi;
    v8f c = {0.f, 0.f, 0.f, 0.f, 0.f, 0.f, 0.f, 0.f};
    #pragma unroll
    for (int kk = 0; kk < 32; ++kk) {
      v2f b = *(const v2f*)(bp + kk * 4);
      c = __builtin_amdgcn_wmma_f32_16x16x4_f32(false, a[kk], false, b,
                                                (short)0, c, false, false);
    }
    const float bv  = bias[col_base + n];
    const int   col = col_base + n;
    #pragma unroll
    for (int r = 0; r < 8; ++r) {
      const int row = rb * 16 + r + 8 * hi;
      if (row < L) out[(size_t)row * VOCAB + col] = c[r] + bv;
    }
  }
}

// ---------------------------------------------------------------------------
// Kernel 5: per-row online (max, sum-exp). One block per row.
// ---------------------------------------------------------------------------
__global__ __launch_bounds__(256) void k_rowstats(const float* __restrict__ logits,
                                                  float* __restrict__ rowM,
                                                  float* __restrict__ rowS) {
  __shared__ float sm[256], ss[256];
  const int row = blockIdx.x;
  const int tid = threadIdx.x;
  const float* p = logits + (size_t)row * VOCAB;
  float m = -1e30f, s = 0.f;
  for (int i = tid; i < VOCAB; i += 256) {
    const float v = p[i];
    const float M = fmaxf(m, v);
    s = fmaf(s, __expf(m - M), __expf(v - M));
    m = M;
  }
  sm[tid] = m; ss[tid] = s;
  __syncthreads();
  for (int off = 128; off > 0; off >>= 1) {
    if (tid < off) {
      const float m2 = sm[tid + off], s2 = ss[tid + off];
      const float M = fmaxf(sm[tid], m2);
      ss[tid] = ss[tid] * __expf(sm[tid] - M) + s2 * __expf(m2 - M);
      sm[tid] = M;
    }
    __syncthreads();
  }
  if (tid == 0) { rowM[row] = sm[0]; rowS[row] = ss[0]; }
}

// ---------------------------------------------------------------------------
// Kernel 6: normalize in place: out = exp(out - rowM) / rowS
// ---------------------------------------------------------------------------
__global__ __launch_bounds__(256) void k_norm(float* __restrict__ out,
                                              const float* __restrict__ rowM,
                                              const float* __restrict__ rowS) {
  const size_t total = (size_t)L * VOCAB;
  const size_t i = (size_t)blockIdx.x * blockDim.x + threadIdx.x;
  if (i >= total) return;
  const int row = (int)(i >> 14);  // VOCAB == 2^14
  out[i] = __expf(out[i] - rowM[row]) * (1.f / rowS[row]);
}

// ---------------------------------------------------------------------------
extern "C" void kernel_launch(void* const* d_in, const int* in_sizes, int n_in,
                              void* d_out, int out_size, void* d_ws, size_t ws_size,
                              hipStream_t stream) {
  const int*   codes  = (const int*)  d_in[0];
  const int*   visits = (const int*)  d_in[1];
  const float* emb    = (const float*)d_in[2];
  const float* w_ih   = (const float*)d_in[3];
  const float* w_hh   = (const float*)d_in[4];
  const float* b_ih   = (const float*)d_in[5];
  const float* b_hh   = (const float*)d_in[6];
  const float* w_out  = (const float*)d_in[7];
  const float* b_out  = (const float*)d_in[8];
  float* out = (float*)d_out;

  // workspace layout (floats)
  float* ws       = (float*)d_ws;
  float* sum_emb  = ws;                              // VOCAB
  float* scalars  = sum_emb + VOCAB;                 // SEG
  float* gru_out  = scalars + SEG;                   // MPAD * H
  float* rowM     = gru_out + (size_t)MPAD * H;      // L
  float* rowS     = rowM + L;                        // L

  // 1) emb row sums: one wave per vocab row
  k_embsum<<<(VOCAB * 32 + 255) / 256, 256, 0, stream>>>(emb, sum_emb);
  // 2) segment scalars
  k_segsum<<<(SEG + 255) / 256, 256, 0, stream>>>(codes, visits, sum_emb, scalars);
  // 3) serial GRU scan (single workgroup)
  k_gru<<<1, 128, 0, stream>>>(scalars, w_ih, w_hh, b_ih, b_hh, gru_out);
  // 4) WMMA fp32 GEMM + bias -> raw logits in d_out
  k_gemm<<<MTILES, 256, 0, stream>>>(gru_out, w_out, b_out, out);
  // 5) per-row softmax statistics
  k_rowstats<<<L, 256, 0, stream>>>(out, rowM, rowS);
  // 6) normalize
  const size_t total = (size_t)L * VOCAB;
  k_norm<<<(unsigned)((total + 255) / 256), 256, 0, stream>>>(out, rowM, rowS);
}